// MultiInputLSTMCell_V0_40261023432818
// MI455X (gfx1250) — compile-verified
//
#include <hip/hip_runtime.h>
#include <cmath>

typedef __attribute__((ext_vector_type(2))) float v2f;
typedef __attribute__((ext_vector_type(4))) float v4f;
typedef __attribute__((ext_vector_type(8))) float v8f;
typedef __attribute__((ext_vector_type(4))) int   v4i;

#define B_TOT 16384
#define XCNT  50
#define CINW  128
#define HW    128

#define SKIP_LD 132   // padded row stride (floats) for skip_c tiles in LDS
#define WHH_LD  136   // padded row stride (floats) for alpha_weight_hh in LDS

// ---- CDNA5 async global->LDS path (graceful fallback if builtin names differ) ----
#if __has_builtin(__builtin_amdgcn_global_load_async_to_lds_b128) && \
    __has_builtin(__builtin_amdgcn_s_wait_asynccnt)
#define HAVE_ASYNC_LDS 1
typedef __attribute__((address_space(1))) v4i as1_v4i;   // global int4*
typedef __attribute__((address_space(3))) v4i as3_v4i;   // LDS int4*
#else
#define HAVE_ASYNC_LDS 0
#endif

__device__ __forceinline__ float fast_rcp(float x) { return __builtin_amdgcn_rcpf(x); }
__device__ __forceinline__ float sigmoidf_(float x) { return fast_rcp(1.0f + __expf(-x)); }
// branch-free tanh: 1 - 2/(e^{2x}+1); correct limits at +/-inf
__device__ __forceinline__ float fast_tanh(float x) {
    return 1.0f - 2.0f * fast_rcp(__expf(2.0f * x) + 1.0f);
}

__device__ __forceinline__ v8f wmma_f32_k4(v2f a, v2f b, v8f c) {
    // D = A(16x4 f32) * B(4x16 f32) + C(16x16 f32)
    return __builtin_amdgcn_wmma_f32_16x16x4_f32(false, a, false, b, (short)0, c, false, false);
}

__device__ __forceinline__ void stage_tile(float* __restrict__ dst,
                                           const float* __restrict__ skip_c,
                                           int r0, int x, int pairLane) {
    // wave pair (64 lanes) stages skip_c[r0..r0+15, x, 0..127] (8 KB) into LDS
    #pragma unroll
    for (int q = 0; q < 8; ++q) {
        int linear = q * 64 + pairLane;
        int row = linear >> 5, c4 = (linear & 31) << 2;
        const float* src = &skip_c[((size_t)(r0 + row) * XCNT + x) * HW + c4];
        float* d = &dst[row * SKIP_LD + c4];
#if HAVE_ASYNC_LDS
        __builtin_amdgcn_global_load_async_to_lds_b128(
            (as1_v4i*)src, (as3_v4i*)d, 0, 0);
#else
        *(v4f*)d = *(const v4f*)src;
#endif
    }
}

__global__ __launch_bounds__(256) void milstm_fused_kernel(
    const float* __restrict__ inp,   const float* __restrict__ skip_c,
    const int*   __restrict__ skip_count, const float* __restrict__ h0,
    const float* __restrict__ Wih,   const float* __restrict__ Whh,
    const float* __restrict__ bias,  const float* __restrict__ aWih,
    const float* __restrict__ aWhh,  const float* __restrict__ abias,
    float* __restrict__ out_h, float* __restrict__ out_c)
{
    __shared__ alignas(16) float lds_whh[128 * WHH_LD];            // ~68 KB
    __shared__ alignas(16) float lds_skip[4][2][16 * SKIP_LD];     // double-buffered, ~66 KB

    const int tid   = threadIdx.x;
    const int lane  = tid & 31;
    const int wave  = tid >> 5;
    const int t     = wave >> 1;          // batch sub-tile 0..3 (16 rows each)
    const int hf    = wave & 1;           // column half of H
    const int row16 = lane & 15;
    const int hi    = lane >> 4;          // 0 or 1 (C-layout M offset 8*hi)
    const int khalf = hi << 1;            // K sub-offset 0 or 2 for A/B frags

    const int r0 = blockIdx.x * 64 + t * 16;  // first batch row of this wave's tile
    const int nb = hf * 64;                   // first H column of this wave
    const int pairLane = tid & 63;

    // ---- stage alpha_weight_hh [128,128] -> LDS (padded, conflict-free B-frag reads)
    for (int idx = tid; idx < 128 * 32; idx += 256) {
        int r = idx >> 5, c4 = (idx & 31) << 2;
        *(v4f*)&lds_whh[r * WHH_LD + c4] = *(const v4f*)&aWhh[r * 128 + c4];
    }

    // kick off the first skip_c tile DMA while we do phase 1
    stage_tile(&lds_skip[t][0][0], skip_c, r0, 0, pairLane);

    // ---- phase 1: gates (i,o,g) and alpha input pre-activation via f32 WMMA ----
    // acc[m*4+nt]: m=0 -> e_i, 1 -> o, 2 -> g, 3 (12..15) -> a_in; nt = N-tile 0..3
    v8f acc[16];
    #pragma unroll
    for (int m = 0; m < 16; ++m)
        acc[m] = (v8f){0.f,0.f,0.f,0.f,0.f,0.f,0.f,0.f};

    for (int k = 0; k < 32; ++k) {
        const int brow = 4 * k + khalf;
        v2f aI = *(const v2f*)&inp[(r0 + row16) * CINW + brow];
        v2f aH = *(const v2f*)&h0 [(r0 + row16) * HW   + brow];
        #pragma unroll
        for (int nt = 0; nt < 4; ++nt) {
            const int col = nb + nt * 16 + row16;
            #pragma unroll
            for (int m = 0; m < 3; ++m) {
                v2f b1, b2;
                b1.x = Wih[ brow      * 384 + m * 128 + col];
                b1.y = Wih[(brow + 1) * 384 + m * 128 + col];
                acc[m * 4 + nt] = wmma_f32_k4(aI, b1, acc[m * 4 + nt]);
                b2.x = Whh[ brow      * 384 + m * 128 + col];
                b2.y = Whh[(brow + 1) * 384 + m * 128 + col];
                acc[m * 4 + nt] = wmma_f32_k4(aH, b2, acc[m * 4 + nt]);
            }
            v2f b3;
            b3.x = aWih[ brow      * 128 + col];
            b3.y = aWih[(brow + 1) * 128 + col];
            acc[12 + nt] = wmma_f32_k4(aI, b3, acc[12 + nt]);
        }
    }

    // activations in the C-tile layout: element (M = r + 8*hi, N = lane&15)
    #pragma unroll
    for (int nt = 0; nt < 4; ++nt) {
        const int col = nb + nt * 16 + row16;
        const float bi = bias[col], bo = bias[128 + col], bg = bias[256 + col], ba = abias[col];
        #pragma unroll
        for (int r = 0; r < 8; ++r) {
            float iv = sigmoidf_(acc[nt][r] + bi);
            acc[nt][r]      = __expf(iv);                       // e_i
            acc[4 + nt][r]  = sigmoidf_(acc[4 + nt][r] + bo);   // o
            acc[8 + nt][r]  = fast_tanh(acc[8 + nt][r] + bg);   // g
            acc[12 + nt][r] = acc[12 + nt][r] + ba;             // a_in (pre-sigmoid)
        }
    }

    int scnt[8];
    #pragma unroll
    for (int r = 0; r < 8; ++r) scnt[r] = skip_count[r0 + r + 8 * hi];

    v8f num[4], den[4];
    #pragma unroll
    for (int nt = 0; nt < 4; ++nt) {
        num[nt] = (v8f){0.f,0.f,0.f,0.f,0.f,0.f,0.f,0.f};
        den[nt] = (v8f){0.f,0.f,0.f,0.f,0.f,0.f,0.f,0.f};
    }

#if HAVE_ASYNC_LDS
    __builtin_amdgcn_s_wait_asynccnt(0);
#endif
    __syncthreads();  // whh + tile(x=0) staged

    // ---- phase 2: stream skip_c once; double-buffered DMA overlaps WMMA + exp-norm ----
    for (int x = 0; x < XCNT; ++x) {
        float* cur = &lds_skip[t][x & 1][0];

        if (x + 1 < XCNT) {
            // issue next tile's async DMA into the other buffer (overlaps compute)
            stage_tile(&lds_skip[t][(x + 1) & 1][0], skip_c, r0, x + 1, pairLane);
#if !HAVE_ASYNC_LDS
            __builtin_prefetch(
                &skip_c[((size_t)(r0 + (pairLane >> 2)) * XCNT + x + 2) * HW + (pairLane & 3) * 32],
                0, 0);
#endif
        }

        v8f s[4];
        #pragma unroll
        for (int nt = 0; nt < 4; ++nt)
            s[nt] = (v8f){0.f,0.f,0.f,0.f,0.f,0.f,0.f,0.f};

        #pragma unroll
        for (int k = 0; k < 32; ++k) {
            const int brow = 4 * k + khalf;
            v2f a = *(const v2f*)&cur[row16 * SKIP_LD + brow];
            #pragma unroll
            for (int nt = 0; nt < 4; ++nt) {
                const int col = nb + nt * 16 + row16;
                v2f b;
                b.x = lds_whh[ brow      * WHH_LD + col];
                b.y = lds_whh[(brow + 1) * WHH_LD + col];
                s[nt] = wmma_f32_k4(a, b, s[nt]);
            }
        }

        #pragma unroll
        for (int nt = 0; nt < 4; ++nt) {
            const int col = nb + nt * 16 + row16;
            #pragma unroll
            for (int r = 0; r < 8; ++r) {
                float alpha = sigmoidf_(acc[12 + nt][r] + s[nt][r]);
                float e  = (x < scnt[r]) ? __expf(alpha) : 0.0f;
                float sv = cur[(r + 8 * hi) * SKIP_LD + col];
                num[nt][r] += sv * e;
                den[nt][r] += e;
            }
        }

#if HAVE_ASYNC_LDS
        __builtin_amdgcn_s_wait_asynccnt(0);  // my x+1 DMA landed
#endif
        __syncthreads();                      // everyone's landed; cur free for x+2
    }

    // ---- finalize: c1 = (g*e_i + num) / (e_i + den); h1 = o * tanh(c1) ----
    #pragma unroll
    for (int nt = 0; nt < 4; ++nt) {
        const int col = nb + nt * 16 + row16;
        #pragma unroll
        for (int r = 0; r < 8; ++r) {
            const int b = r0 + r + 8 * hi;
            float e_i = acc[nt][r];
            float c1  = (acc[8 + nt][r] * e_i + num[nt][r]) * fast_rcp(e_i + den[nt][r]);
            float h1  = acc[4 + nt][r] * fast_tanh(c1);
            out_h[(size_t)b * HW + col] = h1;
            out_c[(size_t)b * HW + col] = c1;
        }
    }
}

extern "C" void kernel_launch(void* const* d_in, const int* in_sizes, int n_in,
                              void* d_out, int out_size, void* d_ws, size_t ws_size,
                              hipStream_t stream) {
    const float* inp   = (const float*)d_in[0];
    const float* skip  = (const float*)d_in[1];
    const int*   scnt  = (const int*)  d_in[2];
    const float* h0    = (const float*)d_in[3];
    // d_in[4] = c0 : unused by the reference
    const float* Wih   = (const float*)d_in[5];
    const float* Whh   = (const float*)d_in[6];
    const float* bias  = (const float*)d_in[7];
    const float* aWih  = (const float*)d_in[8];
    const float* aWhh  = (const float*)d_in[9];
    const float* abias = (const float*)d_in[10];

    float* out_h = (float*)d_out;                      // [B, H]
    float* out_c = out_h + (size_t)B_TOT * HW;         // [B, H]

    dim3 grid(B_TOT / 64), block(256);
    hipLaunchKernelGGL(milstm_fused_kernel, grid, block, 0, stream,
                       inp, skip, scnt, h0, Wih, Whh, bias, aWih, aWhh, abias,
                       out_h, out_c);
}